// PWCoordAttention_49976239456478
// MI455X (gfx1250) — compile-verified
//
#include <hip/hip_runtime.h>
#include <math.h>
#include <stdint.h>

typedef __attribute__((ext_vector_type(2))) float v2f;
typedef __attribute__((ext_vector_type(8))) float v8f;

#define BB   128
#define CC   256
#define NC   64
#define TC   8
#define HH   32
#define WW   32
#define NPIX 1024   // 32*32
#define EPSV 1e-5f

// ---------------- workspace layout (floats) ----------------
// corr : [B][64][1024]              at 0                (8,388,608)
// xcat : [B][64][64]                at 8,388,608        (524,288)
// y    : [B][8][64]                 at 8,912,896        (65,536)
// stats: mu[8], rstd[8]             at 8,978,432        (16)
#define WS_CORR  0
#define WS_XCAT  8388608
#define WS_Y     8912896
#define WS_STATS 8978432

// =====================================================================
// K1: corr[b] = z[b]^T (64x256) @ x[b] (256x1024) via v_wmma_f32_16x16x4_f32.
// grid (B, 4 N-strips of 256) ; block 256 (8 waves).
// Block covers ALL of M=64 for its N-strip, so each x element enters the
// WGP exactly once: the 4x256 fp32 K-chunk is staged into LDS with
// global_load_async_to_lds_b128 (ASYNCcnt) and shared by all 8 waves.
// Wave w: M-tile = (w&3)*16, N-half = (w>>2)*128 (8 WMMA tiles of 16x16).
// =====================================================================
__global__ __launch_bounds__(256)
void k1_corr_wmma(const float* __restrict__ z, const float* __restrict__ x,
                  float* __restrict__ ws) {
    __shared__ __attribute__((aligned(16))) float xs[4 * 256]; // K-chunk [k'][n]

    const int b    = blockIdx.x;
    const int nblk = blockIdx.y * 256;           // N base of this block
    const int tid  = threadIdx.x;
    const int wave = tid >> 5;
    const int lane = tid & 31;
    const int m16  = lane & 15;
    const int g    = lane >> 4;                  // K half (0: K=0,1 ; 1: K=2,3)
    const int k0   = (wave & 3) * 16;            // M tile of this wave
    const int nh   = (wave >> 2) * 128;          // N half within the strip

    const float* Zb = z + (size_t)b * CC * NC;
    const float* Xb = x + (size_t)b * CC * NPIX;

    // staging: thread -> (row sr of the K-chunk, 4 consecutive cols)
    const int sr = tid >> 6;                     // 0..3
    const int sc = (tid & 63) * 4;               // 0,4,...,252
    const float* stage_src = Xb + (size_t)sr * NPIX + nblk + sc;
    const unsigned lds_dst = (unsigned)(uintptr_t)(&xs[sr * 256 + sc]);

    v8f acc[8];
#pragma unroll
    for (int t = 0; t < 8; ++t) acc[t] = (v8f){0,0,0,0,0,0,0,0};

    const int za = (2 * g) * NC + k0 + m16;      // A[m][c] = z[c][m]

    for (int c0 = 0; c0 < CC; c0 += 4) {
        // async-copy this K-chunk of x into LDS (16B per lane = 4KB/block)
        asm volatile("global_load_async_to_lds_b128 %0, %1, off"
                     :: "v"(lds_dst), "v"(stage_src + (size_t)c0 * NPIX)
                     : "memory");
        // A fragment from global (z is small & L2-resident)
        v2f af;
        af.x = Zb[c0 * NC + za];                 // K = c0+2g
        af.y = Zb[c0 * NC + za + NC];            // K = c0+2g+1
        asm volatile("s_wait_asynccnt 0x0" ::: "memory");
        __syncthreads();
#pragma unroll
        for (int t = 0; t < 8; ++t) {
            v2f bf;
            bf.x = xs[(2 * g) * 256 + nh + t * 16 + m16];
            bf.y = xs[(2 * g + 1) * 256 + nh + t * 16 + m16];
            acc[t] = __builtin_amdgcn_wmma_f32_16x16x4_f32(
                false, af, false, bf, (short)0, acc[t], false, false);
        }
        __syncthreads();                         // LDS reused next K-chunk
    }

    // store D tiles: VGPR r holds M = k0 + r + 8*g, N = nblk + nh + t*16 + m16
    float* Cb = ws + WS_CORR + ((size_t)b * NC + k0) * NPIX + nblk + nh;
#pragma unroll
    for (int t = 0; t < 8; ++t) {
#pragma unroll
        for (int r = 0; r < 8; ++r) {
            Cb[(size_t)(r + 8 * g) * NPIX + t * 16 + m16] = acc[t][r];
        }
    }
}

// =====================================================================
// K1b: x_cat[b][k][0:32]  = row means of corr[b][k]  (mean over j)
//      x_cat[b][k][32:64] = col means of corr[b][k]  (mean over i)
// one wave per (b,k); lane j owns column j.
// =====================================================================
__global__ __launch_bounds__(32)
void k1b_means(const float* __restrict__ ws_corr, float* __restrict__ ws_xcat) {
    const int bk   = blockIdx.x;                 // 0 .. B*64-1
    const int lane = threadIdx.x;                // 0..31 (column j)
    const float* Cb = ws_corr + (size_t)bk * NPIX;

    float colsum = 0.0f;
    float myrow  = 0.0f;
    for (int i = 0; i < HH; ++i) {
        float v = Cb[i * WW + lane];             // coalesced: 32 lanes = one row
        colsum += v;
        float r = v;                             // reduce row i across lanes
        r += __shfl_xor(r, 16, 32);
        r += __shfl_xor(r, 8, 32);
        r += __shfl_xor(r, 4, 32);
        r += __shfl_xor(r, 2, 32);
        r += __shfl_xor(r, 1, 32);
        if (lane == i) myrow = r;                // lane i keeps rowsum(i)
    }
    float* out = ws_xcat + (size_t)bk * 64;
    out[lane]      = myrow  * (1.0f / 32.0f);    // x_h
    out[32 + lane] = colsum * (1.0f / 32.0f);    // x_w
}

// =====================================================================
// K2: y[b] = conv1_w (8x64) @ x_cat[b] (64x64) + conv1_b
// grid B ; block 512 : thread = o*64 + l
// =====================================================================
__global__ __launch_bounds__(512)
void k2_conv1(const float* __restrict__ ws_xcat, const float* __restrict__ w1,
              const float* __restrict__ b1, float* __restrict__ ws_y) {
    const int b = blockIdx.x;
    const int o = threadIdx.x >> 6;
    const int l = threadIdx.x & 63;
    const float* Xc = ws_xcat + (size_t)b * 4096;
    float acc = b1[o];
    for (int c = 0; c < 64; ++c)
        acc += w1[o * 64 + c] * Xc[c * 64 + l];
    ws_y[(size_t)b * 512 + o * 64 + l] = acc;
}

// =====================================================================
// K3: BN stats. single block, 1024 threads: channel = t>>7, 128 partials each.
// =====================================================================
__global__ __launch_bounds__(1024)
void k3_bnstats(const float* __restrict__ ws_y, float* __restrict__ ws_stats) {
    __shared__ float s1[1024];
    __shared__ float s2[1024];
    const int t   = threadIdx.x;
    const int ch  = t >> 7;                      // 0..7
    const int sub = t & 127;
    float sum = 0.0f, ssq = 0.0f;
    for (int idx = sub; idx < BB * 64; idx += 128) {
        int b = idx >> 6, l = idx & 63;
        float v = ws_y[(size_t)b * 512 + ch * 64 + l];
        sum += v; ssq += v * v;
    }
    s1[t] = sum; s2[t] = ssq;
    __syncthreads();
    for (int off = 64; off > 0; off >>= 1) {
        if (sub < off) { s1[t] += s1[t + off]; s2[t] += s2[t + off]; }
        __syncthreads();
    }
    if (sub == 0) {
        const float n  = (float)(BB * 64);
        float mu  = s1[t] / n;
        float var = s2[t] / n - mu * mu;
        ws_stats[ch]     = mu;
        ws_stats[8 + ch] = rsqrtf(var + EPSV);
    }
}

// =====================================================================
// K4: per (b, kgroup of 8): BN+h_swish(y[b]) -> gates -> out = corr*ow*oh
// grid (B, 8) ; block 256
// =====================================================================
__device__ __forceinline__ float hswish(float v) {
    return v * fminf(fmaxf(v + 3.0f, 0.0f), 6.0f) * (1.0f / 6.0f);
}
__device__ __forceinline__ float sigmoidf(float v) {
    return 1.0f / (1.0f + expf(-v));
}

__global__ __launch_bounds__(256)
void k4_gate_apply(const float* __restrict__ ws, const float* __restrict__ gamma,
                   const float* __restrict__ beta, const float* __restrict__ w2,
                   const float* __restrict__ b2, const float* __restrict__ w3,
                   const float* __restrict__ b3, float* __restrict__ out) {
    __shared__ float yh[TC * 32];    // h-swished BN'd y, first 32 positions
    __shared__ float yw[TC * 32];    // last 32 positions
    __shared__ float oh[8 * 32];     // sigmoid gate (rows)  for this kgroup
    __shared__ float ow[8 * 32];     // sigmoid gate (cols)

    const int b   = blockIdx.x;
    const int kg  = blockIdx.y;      // 8 k-channels per block
    const int tid = threadIdx.x;

    const float* ws_y     = ws + WS_Y;
    const float* ws_stats = ws + WS_STATS;

    // BN + h-swish of y[b] (8x64) -> LDS
    for (int idx = tid; idx < TC * 64; idx += 256) {
        int o = idx >> 6, l = idx & 63;
        float v = ws_y[(size_t)b * 512 + idx];
        v = (v - ws_stats[o]) * ws_stats[8 + o] * gamma[o] + beta[o];
        v = hswish(v);
        if (l < 32) yh[o * 32 + l] = v;
        else        yw[o * 32 + (l - 32)] = v;
    }
    __syncthreads();

    // gates for k = kg*8 .. kg*8+7 : one (kk, pos) per thread
    {
        int kk = tid >> 5, p = tid & 31;
        int k  = kg * 8 + kk;
        float sh = b2[k], sw = b3[k];
#pragma unroll
        for (int c = 0; c < TC; ++c) {
            sh += w2[k * TC + c] * yh[c * 32 + p];
            sw += w3[k * TC + c] * yw[c * 32 + p];
        }
        oh[kk * 32 + p] = sigmoidf(sh);
        ow[kk * 32 + p] = sigmoidf(sw);
    }
    __syncthreads();

    // streaming apply: 8 k-channels * 1024 pixels = 8192 elems, 32 iters
    const float* corr = ws + WS_CORR + ((size_t)b * NC + kg * 8) * NPIX;
    float* outb       = out + ((size_t)b * NC + kg * 8) * NPIX;
    for (int it = 0; it < 32; ++it) {
        int idx = it * 256 + tid;                // 0..8191
        int kk  = idx >> 10;
        int rem = idx & 1023;
        int i   = rem >> 5;
        int j   = rem & 31;
        outb[idx] = corr[idx] * ow[kk * 32 + j] * oh[kk * 32 + i];
    }
}

// =====================================================================
extern "C" void kernel_launch(void* const* d_in, const int* in_sizes, int n_in,
                              void* d_out, int out_size, void* d_ws, size_t ws_size,
                              hipStream_t stream) {
    const float* z  = (const float*)d_in[0];
    const float* x  = (const float*)d_in[1];
    const float* w1 = (const float*)d_in[2];
    const float* b1 = (const float*)d_in[3];
    const float* gm = (const float*)d_in[4];
    const float* bt = (const float*)d_in[5];
    const float* w2 = (const float*)d_in[6];
    const float* b2 = (const float*)d_in[7];
    const float* w3 = (const float*)d_in[8];
    const float* b3 = (const float*)d_in[9];
    float* ws  = (float*)d_ws;
    float* out = (float*)d_out;

    k1_corr_wmma<<<dim3(BB, 4), 256, 0, stream>>>(z, x, ws);
    k1b_means<<<BB * NC, 32, 0, stream>>>(ws + WS_CORR, ws + WS_XCAT);
    k2_conv1<<<BB, 512, 0, stream>>>(ws + WS_XCAT, w1, b1, ws + WS_Y);
    k3_bnstats<<<1, 1024, 0, stream>>>(ws + WS_Y, ws + WS_STATS);
    k4_gate_apply<<<dim3(BB, 8), 256, 0, stream>>>(ws, gm, bt, w2, b2, w3, b3, out);
}